// StreamingRhythmProjector_66331474919804
// MI455X (gfx1250) — compile-verified
//
#include <hip/hip_runtime.h>
#include <cstdint>

#define T_LEN   8192
#define B_ROWS  512
#define KEEP_K  2867          // max(1, int(round(8192*0.35)))
#define NT      1024          // threads per block (32 waves, wave32)
#define NH      16            // replicated histogram copies

#define MIN_SPEECH 1.0f
#define MAX_EXPAND 3.0f
#define MIN_BW     0.1f
#define BIAS_W     0.15f
#define INV_TEMP   (1.0f / 0.12f)

#if defined(__has_builtin)
#if __has_builtin(__builtin_amdgcn_tensor_load_to_lds) && \
    __has_builtin(__builtin_amdgcn_s_wait_tensorcnt)
#define USE_TDM 1
#endif
#if __has_builtin(__builtin_amdgcn_global_load_async_to_lds_b128) && \
    __has_builtin(__builtin_amdgcn_s_wait_asynccnt)
#define USE_ASYNC_LDS 1
#endif
#endif
#ifndef USE_TDM
#define USE_TDM 0
#endif
#ifndef USE_ASYNC_LDS
#define USE_ASYNC_LDS 0
#endif

#if USE_ASYNC_LDS
typedef int v4i __attribute__((vector_size(16)));
#define GLB_V4I(p) ((v4i*)(uintptr_t)(p))
#define LDS_V4I(p) ((__attribute__((address_space(3))) v4i*)(uint32_t)(uintptr_t)(p))
#endif

#if USE_TDM
typedef unsigned int u32x4 __attribute__((vector_size(16)));
typedef int          i32x4 __attribute__((vector_size(16)));
typedef int          i32x8 __attribute__((vector_size(32)));

// Tensor DMA: copy one contiguous row of T_LEN f32 (32 KB) from global to LDS.
// D# per CDNA5 ISA ch.8: group0 = {count|flags, lds_addr, gaddr_lo, gaddr_hi|type=2},
// group1 = {data_size=4B, tensor_dim0=8192, tensor_dim1=1, tile_dim0=8192,
//           tile_dim1=1, dim0_stride=8192, 0, 0}. Groups 2/3 unused (<=2D tensor).
__device__ __forceinline__ void tdm_load_row_to_lds(const float* gsrc, void* ldsDst) {
    const uint64_t ga = (uint64_t)(uintptr_t)gsrc;
    const uint32_t la = (uint32_t)(uintptr_t)ldsDst;  // generic low 32 bits = LDS offset
    u32x4 g0 = { 1u,                                   // count=1 (valid), user mode
                 la,                                   // lds_addr (bytes)
                 (uint32_t)(ga & 0xFFFFFFFFu),         // global_addr[31:0]
                 (uint32_t)((ga >> 32) & 0x01FFFFFFu)  // global_addr[56:32]
                     | (2u << 30) };                   // type=2 ("image")
    i32x8 g1 = { (int)(2u << 16),                      // data_size=2 -> 4 bytes
                 (int)((unsigned)T_LEN << 16),         // tensor_dim0[15:0] in [31:16]
                 (int)(1u << 16),                      // tensor_dim1 = 1
                 (int)((unsigned)T_LEN << 16),         // tile_dim0 = 8192 in [127:112]
                 1,                                    // tile_dim1 = 1 (tile_dim2 = 0)
                 T_LEN,                                // tensor_dim0_stride[31:0]
                 0, 0 };                               // stride hi, dim1_stride = 0
    i32x4 gz = { 0, 0, 0, 0 };
#if __clang_major__ >= 23
    i32x8 gz8 = { 0, 0, 0, 0, 0, 0, 0, 0 };
    __builtin_amdgcn_tensor_load_to_lds(g0, g1, gz, gz, gz8, 0);   // 6-arg toolchain
#else
    __builtin_amdgcn_tensor_load_to_lds(g0, g1, gz, gz, 0);        // 5-arg toolchain
#endif
}
#endif

__device__ __forceinline__ float waveReduceSum(float v) {
    v += __shfl_down(v, 16, 32);
    v += __shfl_down(v, 8, 32);
    v += __shfl_down(v, 4, 32);
    v += __shfl_down(v, 2, 32);
    v += __shfl_down(v, 1, 32);
    return v;
}

__global__ __launch_bounds__(NT, 1)
void rhythm_projector_kernel(const float* __restrict__ durA,
                             const float* __restrict__ durLR,
                             const float* __restrict__ pauseW,
                             const float* __restrict__ boundL,
                             const float* __restrict__ unitM,
                             const float* __restrict__ speechBud,
                             const float* __restrict__ pauseBud,
                             const float* __restrict__ prevS,
                             const float* __restrict__ prevP,
                             const int*   __restrict__ frontier,
                             float* __restrict__ out) {
    __shared__ __align__(16) float s_scores[T_LEN];   // scores, then tail_cand (pass C overwrites)
    __shared__ __align__(16) float s_cand[T_LEN];
    __shared__ __align__(16) float s_mask[T_LEN];
    __shared__ __align__(16) float s_prevS[T_LEN];
    __shared__ __align__(16) float s_prevP[T_LEN];
    __shared__ unsigned s_hist[NH][256];
    __shared__ float s_acc[8];     // 0 tailSum, 1 candTail, 2 prefS, 3 prefP, 4 denom
    __shared__ float s_bcast[4];   // 0 scale_s, 1 fbcoef, 2 scale_p, 3 rem_p
    __shared__ unsigned s_sel[2];

    const int tid = threadIdx.x;
    const int b   = blockIdx.x;
    const int f   = frontier[b];

    const size_t rowOff = (size_t)b * T_LEN;
    const float* rDurA  = durA  + rowOff;
    const float* rDurLR = durLR + rowOff;
    const float* rPW    = pauseW + rowOff;
    const float* rBL    = boundL + rowOff;
    const float* rMask  = unitM + rowOff;
    const float* rPrevS = prevS + rowOff;
    const float* rPrevP = prevP + rowOff;
    float* outS = out + rowOff;
    float* outP = out + (size_t)B_ROWS * T_LEN + rowOff;

    if (tid < 8) s_acc[tid] = 0.0f;

    // Warm L2 for the streamed rows (gfx1250 global_prefetch_b8).
    if (tid < 256) {
        const int off = tid << 5;   // 256 threads x 128B cachelines = 32KB row
        __builtin_prefetch(rDurA  + off, 0, 0);
        __builtin_prefetch(rDurLR + off, 0, 0);
        __builtin_prefetch(rPW    + off, 0, 0);
        __builtin_prefetch(rBL    + off, 0, 0);
        __builtin_prefetch(rMask  + off, 0, 0);
    }

#if USE_TDM
    // Tensor Data Mover: one DMA instruction per 32KB row, overlapped with the
    // compute streaming pass. Only wave 0 enters this block (TENSOR ops ignore
    // EXEC, so other waves must branch around, not mask).
    if (tid == 0) {
        tdm_load_row_to_lds(rPrevS, (void*)s_prevS);
        tdm_load_row_to_lds(rPrevP, (void*)s_prevP);
    }
#elif USE_ASYNC_LDS
    // Async DMA fallback: 16B per lane, lane-contiguous -> 512B per wave issue.
    for (int i = tid * 4; i < T_LEN; i += NT * 4) {
        __builtin_amdgcn_global_load_async_to_lds_b128(GLB_V4I(rPrevS + i),
                                                       LDS_V4I(&s_prevS[i]), 0, 0);
        __builtin_amdgcn_global_load_async_to_lds_b128(GLB_V4I(rPrevP + i),
                                                       LDS_V4I(&s_prevP[i]), 0, 0);
    }
#endif

    // ---- Pass A: stream the 5 remaining arrays once; build scores/cand/mask in LDS ----
    float aTail = 0.0f, aCand = 0.0f;
    for (int i = tid; i < T_LEN; i += NT) {
        float m   = rMask[i];
        float da  = rDurA[i];
        float dl  = rDurLR[i];
        float pw  = rPW[i];
        float blv = rBL[i];
#if !USE_TDM && !USE_ASYNC_LDS
        s_prevS[i] = rPrevS[i];
        s_prevP[i] = rPrevP[i];
#endif
        float tail   = (i < f) ? 0.0f : m;
        float anchor = fmaxf(da, MIN_SPEECH);
        float basev  = anchor * __expf(dl);
        float cand   = fminf(fmaxf(basev, MIN_SPEECH), anchor * MAX_EXPAND) * m;
        float sc     = (fmaxf(pw, 0.0f) + BIAS_W * (MIN_BW + fmaxf(blv, 0.0f))) * m;
        s_scores[i] = sc;
        s_cand[i]   = cand;
        s_mask[i]   = m;
        aTail += tail;
        aCand += cand * tail;
    }

#if USE_TDM
    __builtin_amdgcn_s_wait_tensorcnt(0);   // no-op for waves with TENSORcnt==0
#elif USE_ASYNC_LDS
    __builtin_amdgcn_s_wait_asynccnt(0);
#endif
    __syncthreads();   // LDS staging complete; s_acc zeros visible

    // ---- Pass A2: prefix sums from LDS ----
    float aPS = 0.0f, aPP = 0.0f;
    for (int i = tid; i < f; i += NT) {
        float m = s_mask[i];
        aPS += s_prevS[i] * m;
        aPP += s_prevP[i] * m;
    }

    float wv;
    wv = waveReduceSum(aTail); if ((tid & 31) == 0) atomicAdd(&s_acc[0], wv);
    wv = waveReduceSum(aCand); if ((tid & 31) == 0) atomicAdd(&s_acc[1], wv);
    wv = waveReduceSum(aPS);   if ((tid & 31) == 0) atomicAdd(&s_acc[2], wv);
    wv = waveReduceSum(aPP);   if ((tid & 31) == 0) atomicAdd(&s_acc[3], wv);
    __syncthreads();

    if (tid == 0) {
        float tailSum  = s_acc[0];
        float candTail = fmaxf(s_acc[1], 1e-6f);
        float prefS    = s_acc[2];
        float prefP    = s_acc[3];
        float sbud  = fmaxf(speechBud[b], prefS + tailSum * MIN_SPEECH);
        float pbud  = fmaxf(pauseBud[b],  prefP);
        float rem_s = sbud - prefS;
        s_bcast[0] = (tailSum > 0.0f && rem_s > 0.0f) ? rem_s / candTail : 0.0f;
        s_bcast[1] = 1e-6f / fmaxf(tailSum, 1.0f);    // fallback coefficient
        s_bcast[3] = fmaxf(pbud - prefP, 0.0f);        // rem_p
    }
    __syncthreads();

    // ---- Radix select: k-th largest score (k = KEEP_K), 4 rounds of 8 bits ----
    unsigned pref = 0, kRem = KEEP_K;
    for (int rnd = 0; rnd < 4; ++rnd) {
        const int shift = 24 - 8 * rnd;
        for (int i = tid; i < NH * 256; i += NT) ((unsigned*)s_hist)[i] = 0u;
        __syncthreads();
        unsigned* myh = s_hist[(tid >> 6) & (NH - 1)];
        for (int i = tid; i < T_LEN; i += NT) {
            unsigned u = __float_as_uint(s_scores[i]);
            bool sel = (rnd == 0) || ((u >> (shift + 8)) == pref);
            if (sel) atomicAdd(&myh[(u >> shift) & 0xFFu], 1u);
        }
        __syncthreads();
        if (tid == 0) {
            unsigned cum = 0; int bin = 0;
            for (int bi = 255; bi >= 0; --bi) {
                unsigned c = 0;
                for (int h = 0; h < NH; ++h) c += s_hist[h][bi];
                if (cum + c >= kRem) { bin = bi; break; }
                cum += c;
            }
            s_sel[0] = (pref << 8) | (unsigned)bin;
            s_sel[1] = kRem - cum;
        }
        __syncthreads();
        pref = s_sel[0];
        kRem = s_sel[1];
    }
    const float thr = __uint_as_float(pref);

    // ---- Pass C: gated tail candidates + denom reduction (all from LDS) ----
    const float scale_s = s_bcast[0];
    const float fbcoef  = s_bcast[1];
    float dloc = 0.0f;
    for (int i = tid; i < T_LEN; i += NT) {
        float sc   = s_scores[i];
        float m    = s_mask[i];
        float tail = (i < f) ? 0.0f : m;
        float gate = 1.0f / (1.0f + __expf((thr - sc) * INV_TEMP));
        float sparse = sc * gate * m;
        float tc = fmaxf(sparse, 0.0f) * tail + tail * fbcoef;
        s_scores[i] = tc;                    // reuse slot: scores no longer needed
        dloc += tc * tail;
    }
    wv = waveReduceSum(dloc); if ((tid & 31) == 0) atomicAdd(&s_acc[4], wv);
    __syncthreads();

    if (tid == 0) {
        float denom   = fmaxf(s_acc[4], 1e-6f);
        float tailSum = s_acc[0];
        float rem_p   = s_bcast[3];
        s_bcast[2] = (tailSum > 0.0f && rem_p > 0.0f) ? rem_p / denom : 0.0f;
    }
    __syncthreads();

    // ---- Pass D: outputs (write-only to HBM; all operands resident in LDS) ----
    const float scale_p = s_bcast[2];
    for (int i = tid; i < T_LEN; i += NT) {
        float m     = s_mask[i];
        bool  isPre = (i < f);
        float preF  = isPre ? 1.0f : 0.0f;
        float tail  = isPre ? 0.0f : m;
        float sp = (s_prevS[i] * m * preF + s_cand[i]   * scale_s * tail) * m;
        float pa = (s_prevP[i] * m * preF + s_scores[i] * scale_p * tail) * m;
        outS[i] = sp;
        outP[i] = pa;
    }
}

extern "C" void kernel_launch(void* const* d_in, const int* in_sizes, int n_in,
                              void* d_out, int out_size, void* d_ws, size_t ws_size,
                              hipStream_t stream) {
    (void)in_sizes; (void)n_in; (void)out_size; (void)d_ws; (void)ws_size;
    const float* durA      = (const float*)d_in[0];
    const float* durLR     = (const float*)d_in[1];
    const float* pauseW    = (const float*)d_in[2];
    const float* boundL    = (const float*)d_in[3];
    const float* unitM     = (const float*)d_in[4];
    const float* speechBud = (const float*)d_in[5];
    const float* pauseBud  = (const float*)d_in[6];
    const float* prevS     = (const float*)d_in[7];
    const float* prevP     = (const float*)d_in[8];
    const int*   frontier  = (const int*)d_in[9];
    float* out = (float*)d_out;

    rhythm_projector_kernel<<<dim3(B_ROWS), dim3(NT), 0, stream>>>(
        durA, durLR, pauseW, boundL, unitM, speechBud, pauseBud,
        prevS, prevP, frontier, out);
}